// Bispectrum_31576599561124
// MI455X (gfx1250) — compile-verified
//
#include <hip/hip_runtime.h>
#include <math.h>

typedef float v2f __attribute__((ext_vector_type(2)));
typedef float v8f __attribute__((ext_vector_type(8)));

#define NLMAX 5   // l = 0..4
#define NF    8   // features per l
#define MAXT  80

// ---------------------------------------------------------------------------
// Triple metadata, generated at compile time in the exact order the reference
// concatenates blocks: outer L = 0..4 (second combine), inner (l1,l2) lex
// order (first combine's append order per lam).
// ---------------------------------------------------------------------------
struct Meta {
  int nt;
  int l1[MAXT], l2[MAXT], L[MAXT];
  int cg_off[MAXT];
  int out_base[MAXT];
  int cg_total;
  int nout;
};

constexpr Meta build_meta() {
  Meta m = {};
  int cg = 0, base = 0;
  m.nt = 0;
  for (int L = 0; L < NLMAX; ++L) {
    int pair = 0;
    for (int l1 = 0; l1 < NLMAX; ++l1)
      for (int l2 = 0; l2 < NLMAX; ++l2) {
        int lo = l1 > l2 ? l1 - l2 : l2 - l1;
        int hi = l1 + l2; if (hi > 4) hi = 4;
        if (L < lo || L > hi) continue;
        m.l1[m.nt] = l1; m.l2[m.nt] = l2; m.L[m.nt] = L;
        m.cg_off[m.nt] = cg;
        cg += (2*l1+1) * (2*l2+1) * (2*L+1);
        m.out_base[m.nt] = base + pair * (NF*NF*NF); // 64 ps-feat * 8
        ++pair; ++m.nt;
      }
    base += pair * (NF*NF*NF);
  }
  m.cg_total = cg;
  m.nout = base;   // = 33280
  return m;
}

constexpr Meta META = build_meta();
constexpr int NT   = META.nt;        // 65 triples
constexpr int NOUT = META.nout;      // 33280 invariants per batch element
constexpr int CGT  = META.cg_total;  // dense CG table size (~13k floats)

// Per-L contiguous triple ranges (triples are generated L-major).
struct LGroups { int start[NLMAX]; int count[NLMAX]; };
constexpr LGroups build_groups() {
  LGroups g = {};
  for (int L = 0; L < NLMAX; ++L) { g.start[L] = -1; g.count[L] = 0; }
  for (int t = 0; t < META.nt; ++t) {
    int L = META.L[t];
    if (g.start[L] < 0) g.start[L] = t;
    g.count[L]++;
  }
  return g;
}
constexpr LGroups GRP = build_groups();

__constant__ Meta d_meta = build_meta();
__device__ float g_cg[CGT];

// ---------------------------------------------------------------------------
// On-device Clebsch-Gordan coefficients (Racah formula, double precision).
// Built once per launch; deterministic so graph replay is safe.
// ---------------------------------------------------------------------------
__device__ inline double dfact(int n) {
  double r = 1.0;
  for (int i = 2; i <= n; ++i) r *= (double)i;
  return r;
}

__device__ double cg_dev(int j1, int j2, int j3, int m1, int m2, int m3) {
  if (m1 + m2 != m3) return 0.0;
  int lo = j1 > j2 ? j1 - j2 : j2 - j1;
  if (j3 < lo || j3 > j1 + j2) return 0.0;
  double pref = (2.0*j3 + 1.0) * dfact(j3 + j1 - j2) * dfact(j3 - j1 + j2)
              * dfact(j1 + j2 - j3) / dfact(j1 + j2 + j3 + 1);
  pref *= dfact(j3 + m3) * dfact(j3 - m3) * dfact(j1 - m1)
        * dfact(j1 + m1) * dfact(j2 - m2) * dfact(j2 + m2);
  double s = 0.0;
  for (int k = 0; k <= j1 + j2 - j3; ++k) {
    int b1 = j1 + j2 - j3 - k, b2 = j1 - m1 - k, b3 = j2 + m2 - k;
    int b4 = j3 - j2 + m1 + k, b5 = j3 - j1 - m2 + k;
    if (b1 < 0 || b2 < 0 || b3 < 0 || b4 < 0 || b5 < 0) continue;
    double d = dfact(k) * dfact(b1) * dfact(b2) * dfact(b3) * dfact(b4) * dfact(b5);
    s += ((k & 1) ? -1.0 : 1.0) / d;
  }
  return sqrt(pref) * s;
}

__global__ void build_cg_kernel() {
  const int stride = gridDim.x * blockDim.x;
  for (int e = blockIdx.x * blockDim.x + threadIdx.x; e < CGT; e += stride) {
    int t = 0;
    while (t + 1 < d_meta.nt && d_meta.cg_off[t + 1] <= e) ++t;
    const int l1 = d_meta.l1[t], l2 = d_meta.l2[t], L = d_meta.L[t];
    const int n2 = 2*l2 + 1, nL = 2*L + 1;
    const int w  = e - d_meta.cg_off[t];
    const int a  = w % nL;
    const int r  = w / nL;
    const int a2 = r % n2;
    const int a1 = r / n2;
    g_cg[e] = (float)cg_dev(l1, l2, L, a1 - l1, a2 - l2, a - L);
  }
}

// ---------------------------------------------------------------------------
// Per-L group processing. nL / kSteps are compile-time constants so the WMMA
// chain unrolls fully. Four barriers per triple:
//   copy CG->LDS | stage1 -> sP | WMMA -> sOut | coalesced store.
// ---------------------------------------------------------------------------
template<int L>
__device__ __forceinline__ void process_group(
    const float (*sX)[NF][10], const float (*sXt)[16][12],
    float (*sP)[12], float (*sOut)[NF], float* sCG,
    float* __restrict__ outB, int tid) {
  constexpr int nL     = 2*L + 1;
  constexpr int kSteps = (nL + 3) >> 2;

  const int wv   = tid >> 5;           // wave id 0..3 -> M-tile
  const int lane = tid & 31;
  const int m16  = lane & 15;
  const int kr   = (lane >> 4) << 1;   // K sub-offset 0/2 (32-bit operand layout)
  const int p    = tid & 63;           // pair-feature row owned in stage 1
  const int h    = tid >> 6;           // a-parity (0/1)
  const int i    = p >> 3, j2 = p & 7;

#pragma unroll 1
  for (int tt = 0; tt < GRP.count[L]; ++tt) {
    const int t  = GRP.start[L] + tt;
    const int l1 = d_meta.l1[t], l2 = d_meta.l2[t];
    const int n1 = 2*l1 + 1, n2 = 2*l2 + 1;
    const int D  = l1 + l2 - L;

    // ---- copy this triple's dense CG slice into LDS (<= 729 floats) ----
    {
      const float* __restrict__ cgg = g_cg + d_meta.cg_off[t];
      const int sz = n1 * n2 * nL;
      for (int e = tid; e < sz; e += 128) sCG[e] = cgg[e];
    }
    __syncthreads();

    // ---- stage 1: P[p][a] = sum_{a1+a2=a+D} X1[i][a1] X2[j2][a2] CG ----
    {
      const float* X1r = &sX[l1][i][0];
      const float* X2r = &sX[l2][j2][0];
      const int step = (n2 - 1) * nL;
#pragma unroll
      for (int k = 0; k < 6; ++k) {
        const int a = h + 2*k;         // parity-interleaved: balanced halves
        float acc = 0.f;
        if (a < nL) {
          const int s = a + D;
          int lo = s - (n2 - 1); if (lo < 0) lo = 0;
          const int hi2 = (s < n1 - 1) ? s : n1 - 1;
          int idx = (lo * n2 + (s - lo)) * nL + a;
          for (int a1 = lo; a1 <= hi2; ++a1, idx += step)
            acc += X1r[a1] * X2r[s - a1] * sCG[idx];
        }
        sP[p][a] = acc;                // zero K-padding for a >= nL
      }
    }
    __syncthreads();

    // ---- stage 2: [64 x nL] x [nL x 8] via V_WMMA_F32_16X16X4_F32 ----
    {
      v8f c = {};
      const int row = wv * 16 + m16;
#pragma unroll
      for (int ks = 0; ks < kSteps; ++ks) {
        const int k0 = (ks << 2) + kr;
        v2f av, bv;
        av.x = sP[row][k0];      av.y = sP[row][k0 + 1];
        bv.x = sXt[L][m16][k0];  bv.y = sXt[L][m16][k0 + 1];
        c = __builtin_amdgcn_wmma_f32_16x16x4_f32(
                /*neg_a=*/false, av, /*neg_b=*/false, bv,
                /*c_mod=*/(short)0, c, /*reuse_a=*/false, /*reuse_b=*/false);
      }
      // C layout: VGPR r holds M=r (lanes 0-15) / M=8+r (lanes 16-31)
      if (m16 < NF) {
        const int pr = wv * 16 + ((lane >> 4) << 3);
#pragma unroll
        for (int r = 0; r < 8; ++r)
          sOut[pr + r][m16] = c[r];
      }
    }
    __syncthreads();

    // ---- coalesced store: sOut layout == output layout (p*8 + j) ----
    {
      float* __restrict__ op = outB + d_meta.out_base[t];
      const float* so = &sOut[0][0];
#pragma unroll
      for (int e = 0; e < 4; ++e)
        op[e * 128 + tid] = so[e * 128 + tid];
    }
    __syncthreads();
  }
}

// ---------------------------------------------------------------------------
// Main bispectrum kernel: 1 block = 1 batch element, 128 threads = 4 waves.
// ---------------------------------------------------------------------------
__global__ __launch_bounds__(128)
void bispec_kernel(const float* __restrict__ x0, const float* __restrict__ x1,
                   const float* __restrict__ x2, const float* __restrict__ x3,
                   const float* __restrict__ x4, float* __restrict__ out) {
  __shared__ float sX  [NLMAX][NF][10];  // raw covariants, a < 2l+1
  __shared__ float sXt [NLMAX][16][12];  // conjugated/scaled B operand, zero-padded
  __shared__ float sP  [64][12];         // current triple's pair-spectrum tile
  __shared__ float sOut[64][NF];         // staged output tile (coalesced store)
  __shared__ float sCG [9*9*9];          // current triple's dense CG slice
  const int b   = blockIdx.x;
  const int tid = threadIdx.x;

  // warm L2 with the CG table (global_prefetch_b8)
  __builtin_prefetch(g_cg + ((size_t)tid * 32) % CGT, 0, 1);

  // ---- load inputs for this batch element into LDS ----
  const float* xs[NLMAX] = {x0, x1, x2, x3, x4};
  for (int l = 0; l < NLMAX; ++l) {
    const int w = 2*l + 1, nel = NF * w;
    const float* src = xs[l] + (size_t)b * nel;
    for (int e = tid; e < nel; e += 128)
      sX[l][e / w][e % w] = src[e];
  }
  __syncthreads();

  // ---- Xt[L][j][a] = X_L[j][2L-a] * (-1)^a / sqrt(2L+1), zero padded ----
  for (int e = tid; e < NLMAX * 16 * 12; e += 128) {
    const int L = e / (16*12), r = e % (16*12), j = r / 12, a = r % 12;
    float v = 0.f;
    if (j < NF && a < 2*L + 1) {
      const float rs = 1.f / sqrtf((float)(2*L + 1));
      v = sX[L][j][2*L - a] * ((a & 1) ? -rs : rs);
    }
    sXt[L][j][a] = v;
  }
  // visibility of sXt is guaranteed by the first barrier inside process_group

  float* outB = out + (size_t)b * NOUT;
  process_group<0>(sX, sXt, sP, sOut, sCG, outB, tid);
  process_group<1>(sX, sXt, sP, sOut, sCG, outB, tid);
  process_group<2>(sX, sXt, sP, sOut, sCG, outB, tid);
  process_group<3>(sX, sXt, sP, sOut, sCG, outB, tid);
  process_group<4>(sX, sXt, sP, sOut, sCG, outB, tid);
}

// ---------------------------------------------------------------------------
extern "C" void kernel_launch(void* const* d_in, const int* in_sizes, int n_in,
                              void* d_out, int out_size, void* d_ws, size_t ws_size,
                              hipStream_t stream) {
  (void)n_in; (void)out_size; (void)d_ws; (void)ws_size;
  const float* x0 = (const float*)d_in[0];
  const float* x1 = (const float*)d_in[1];
  const float* x2 = (const float*)d_in[2];
  const float* x3 = (const float*)d_in[3];
  const float* x4 = (const float*)d_in[4];
  const int B = in_sizes[0] / NF;   // x0 is [B, 8, 1]

  hipLaunchKernelGGL(build_cg_kernel, dim3(32), dim3(128), 0, stream);
  hipLaunchKernelGGL(bispec_kernel, dim3(B), dim3(128), 0, stream,
                     x0, x1, x2, x3, x4, (float*)d_out);
}